// MolecularGraphEncoder_79972291052146
// MI455X (gfx1250) — compile-verified
//
#include <hip/hip_runtime.h>
#include <hip/hip_bf16.h>
#include <math.h>

// ---------------- problem constants ----------------
#define N_NODES 250000
#define N_EDGES 1000000
#define N_MOLS  10000
#define HEADS   4
#define FH      16
#define HID     64
#define DOUT    256
#define E2      (N_EDGES + N_NODES)   // with self loops
#define NEG_SLOPE 0.2f

typedef float v2f __attribute__((ext_vector_type(2)));
typedef float v8f __attribute__((ext_vector_type(8)));

// order-preserving float<->uint encoding for atomic max on floats
__device__ __forceinline__ unsigned encodeOrd(float f) {
    unsigned b = __float_as_uint(f);
    return (b & 0x80000000u) ? ~b : (b | 0x80000000u);
}
__device__ __forceinline__ float decodeOrd(unsigned u) {
    u = (u & 0x80000000u) ? (u & 0x7fffffffu) : ~u;
    return __uint_as_float(u);
}
#define ENC_NEG_INF 0x007FFFFFu   // encodeOrd(-inf)

// ---------------- generic fills ----------------
__global__ void fill_f32_kernel(float* __restrict__ p, float v, long n) {
    long i = (long)blockIdx.x * blockDim.x + threadIdx.x;
    long stride = (long)gridDim.x * blockDim.x;
    for (; i < n; i += stride) p[i] = v;
}
__global__ void fill_u32_kernel(unsigned* __restrict__ p, unsigned v, long n) {
    long i = (long)blockIdx.x * blockDim.x + threadIdx.x;
    long stride = (long)gridDim.x * blockDim.x;
    for (; i < n; i += stride) p[i] = v;
}

// ---------------- WMMA GEMM: C[M,N] = A[M,K] @ B[K,N] (+bias, optional relu) -------
// One wave32 computes one 16x16 tile, accumulating with v_wmma_f32_16x16x4_f32.
// Requires M%16==0, N%16==0, K%4==0 (true for all call sites).
__global__ void wmma_gemm_kernel(const float* __restrict__ A,
                                 const float* __restrict__ B,
                                 const float* __restrict__ bias,
                                 float* __restrict__ C,
                                 int M, int K, int N, int doRelu) {
    const int wavesPerBlock = blockDim.x >> 5;
    const int wave = threadIdx.x >> 5;
    const int tile = blockIdx.x * wavesPerBlock + wave;
    const int tCols = N >> 4;
    const int nTiles = (M >> 4) * tCols;
    if (tile >= nTiles) return;              // wave-uniform exit
    const int tr = tile / tCols;
    const int tc = tile - tr * tCols;
    const int lane = threadIdx.x & 31;
    const int l16 = lane & 15;
    const int lhi = lane >> 4;               // 0 or 1
    const int row = (tr << 4) + l16;         // A row this lane feeds
    const int col = (tc << 4) + l16;         // output column this lane holds

    v8f acc = {};
    #pragma unroll 4
    for (int k0 = 0; k0 < K; k0 += 4) {
        const int ka = k0 + (lhi << 1);      // K index pair for this lane
        v2f a = *(const v2f*)(A + (size_t)row * K + ka);
        v2f b;
        b.x = B[(size_t)ka * N + col];
        b.y = B[(size_t)(ka + 1) * N + col];
        acc = __builtin_amdgcn_wmma_f32_16x16x4_f32(
            false, a, false, b, (short)0, acc, false, false);
    }

    const float bv = bias ? bias[col] : 0.0f;
    #pragma unroll
    for (int v = 0; v < 8; ++v) {
        float o = acc[v] + bv;
        if (doRelu) o = fmaxf(o, 0.0f);
        C[(size_t)((tr << 4) + v + (lhi << 3)) * N + col] = o;
    }
}

// ---------------- attention logits per node/head ----------------
__global__ void att_alpha_kernel(const float* __restrict__ xh,
                                 const float* __restrict__ a_src,
                                 const float* __restrict__ a_dst,
                                 float* __restrict__ alpha_s,
                                 float* __restrict__ alpha_d) {
    int idx = blockIdx.x * blockDim.x + threadIdx.x;
    if (idx >= N_NODES * HEADS) return;
    const int n = idx >> 2, h = idx & 3;
    const float* xr = xh + (size_t)n * HID + h * FH;
    const float* as = a_src + h * FH;
    const float* ad = a_dst + h * FH;
    float ss = 0.f, sd = 0.f;
    #pragma unroll
    for (int f = 0; f < FH; ++f) { ss += xr[f] * as[f]; sd += xr[f] * ad[f]; }
    alpha_s[idx] = ss;
    alpha_d[idx] = sd;
}

__device__ __forceinline__ void edge_sd(const int* __restrict__ ei, int e, int& s, int& d) {
    if (e < N_EDGES) { s = ei[e]; d = ei[N_EDGES + e]; }
    else             { s = d = e - N_EDGES; }
}

// pass 1: per-dst per-head max of leakyrelu(e)
__global__ void edge_max_kernel(const int* __restrict__ ei,
                                const float* __restrict__ alpha_s,
                                const float* __restrict__ alpha_d,
                                unsigned* __restrict__ emaxU) {
    int idx = blockIdx.x * blockDim.x + threadIdx.x;
    if (idx >= E2 * HEADS) return;
    const int e = idx >> 2, h = idx & 3;
    int s, d; edge_sd(ei, e, s, d);
    float v = alpha_s[s * HEADS + h] + alpha_d[d * HEADS + h];
    v = v > 0.f ? v : NEG_SLOPE * v;
    atomicMax(emaxU + d * HEADS + h, encodeOrd(v));
}

// pass 2: ez = exp(e - emax[dst]); denom[dst] += ez
__global__ void edge_exp_kernel(const int* __restrict__ ei,
                                const float* __restrict__ alpha_s,
                                const float* __restrict__ alpha_d,
                                const unsigned* __restrict__ emaxU,
                                float* __restrict__ ezbuf,
                                float* __restrict__ denom) {
    int idx = blockIdx.x * blockDim.x + threadIdx.x;
    if (idx >= E2 * HEADS) return;
    const int e = idx >> 2, h = idx & 3;
    int s, d; edge_sd(ei, e, s, d);
    float v = alpha_s[s * HEADS + h] + alpha_d[d * HEADS + h];
    v = v > 0.f ? v : NEG_SLOPE * v;
    const float emax = decodeOrd(emaxU[d * HEADS + h]);
    const float ez = expf(v - emax);
    ezbuf[idx] = ez;
    atomicAdd(denom + d * HEADS + h, ez);
}

// pass 3: acc[dst, f] += xh[src, f] * (ez / denom[dst])
__global__ void edge_msg_kernel(const int* __restrict__ ei,
                                const float* __restrict__ xh,
                                const float* __restrict__ ezbuf,
                                const float* __restrict__ denom,
                                float* __restrict__ acc) {
    long idx = (long)blockIdx.x * blockDim.x + threadIdx.x;
    if (idx >= (long)E2 * HID) return;
    const int e = (int)(idx >> 6);
    const int f = (int)(idx & 63);
    const int h = f >> 4;
    int s, d; edge_sd(ei, e, s, d);
    const float ez = ezbuf[(size_t)e * HEADS + h];
    const float den = denom[d * HEADS + h];
    const float alpha = ez / (den + 1e-16f);
    atomicAdd(acc + (size_t)d * HID + f, xh[(size_t)s * HID + f] * alpha);
}

// ---------------- per-node LayerNorm over 64 channels (+bias in, optional relu) ----
__global__ void ln_node_kernel(const float* __restrict__ acc,
                               const float* __restrict__ bias,
                               const float* __restrict__ gamma,
                               const float* __restrict__ beta,
                               float* __restrict__ out, int doRelu) {
    const int node = blockIdx.x * (blockDim.x >> 5) + (threadIdx.x >> 5);
    if (node >= N_NODES) return;
    const int lane = threadIdx.x & 31;
    const float* r = acc + (size_t)node * HID;
    float v0 = r[lane] + bias[lane];
    float v1 = r[lane + 32] + bias[lane + 32];
    float s = v0 + v1;
    for (int o = 16; o; o >>= 1) s += __shfl_xor(s, o, 32);
    const float mu = s * (1.0f / 64.0f);
    const float d0 = v0 - mu, d1 = v1 - mu;
    float q = d0 * d0 + d1 * d1;
    for (int o = 16; o; o >>= 1) q += __shfl_xor(q, o, 32);
    const float rinv = rsqrtf(q * (1.0f / 64.0f) + 1e-5f);
    float o0 = d0 * rinv * gamma[lane] + beta[lane];
    float o1 = d1 * rinv * gamma[lane + 32] + beta[lane + 32];
    if (doRelu) { o0 = fmaxf(o0, 0.f); o1 = fmaxf(o1, 0.f); }
    out[(size_t)node * HID + lane] = o0;
    out[(size_t)node * HID + lane + 32] = o1;
}

// ---------------- segment pooling over molecules ----------------
__global__ void pool_kernel(const float* __restrict__ h,
                            const int* __restrict__ batch_idx,
                            float* __restrict__ hsum,
                            float* __restrict__ cnt) {
    long idx = (long)blockIdx.x * blockDim.x + threadIdx.x;
    if (idx >= (long)N_NODES * HID) return;
    const int n = (int)(idx >> 6);
    const int c = (int)(idx & 63);
    const int b = batch_idx[n];
    atomicAdd(hsum + (size_t)b * HID + c, h[idx]);
    if (c == 0) atomicAdd(cnt + b, 1.0f);
}

__global__ void hp_kernel(const float* __restrict__ hsum,
                          const float* __restrict__ cnt,
                          float* __restrict__ hp) {
    int idx = blockIdx.x * blockDim.x + threadIdx.x;
    if (idx >= N_MOLS * 2 * HID) return;
    const int m = idx >> 7;
    const int c = idx & 127;
    const float inv = 1.0f / fmaxf(cnt[m], 1.0f);
    hp[idx] = (c < HID) ? hsum[(size_t)m * HID + c] * inv
                        : hsum[(size_t)m * HID + (c - HID)];
}

// ---------------- final LayerNorm over 256 channels -> d_out ----------------
__global__ void ln_out_kernel(const float* __restrict__ z,
                              const float* __restrict__ gamma,
                              const float* __restrict__ beta,
                              float* __restrict__ out) {
    const int mol = blockIdx.x * (blockDim.x >> 5) + (threadIdx.x >> 5);
    if (mol >= N_MOLS) return;
    const int lane = threadIdx.x & 31;
    const float* r = z + (size_t)mol * DOUT;
    float v[8];
    float s = 0.f;
    #pragma unroll
    for (int j = 0; j < 8; ++j) { v[j] = r[lane + 32 * j]; s += v[j]; }
    for (int o = 16; o; o >>= 1) s += __shfl_xor(s, o, 32);
    const float mu = s * (1.0f / 256.0f);
    float q = 0.f;
    #pragma unroll
    for (int j = 0; j < 8; ++j) { v[j] -= mu; q += v[j] * v[j]; }
    for (int o = 16; o; o >>= 1) q += __shfl_xor(q, o, 32);
    const float rinv = rsqrtf(q * (1.0f / 256.0f) + 1e-5f);
    #pragma unroll
    for (int j = 0; j < 8; ++j)
        out[(size_t)mol * DOUT + lane + 32 * j] =
            v[j] * rinv * gamma[lane + 32 * j] + beta[lane + 32 * j];
}

// ---------------- host-side launch ----------------
static inline int cdiv(long a, long b) { return (int)((a + b - 1) / b); }

extern "C" void kernel_launch(void* const* d_in, const int* in_sizes, int n_in,
                              void* d_out, int out_size, void* d_ws, size_t ws_size,
                              hipStream_t stream) {
    (void)in_sizes; (void)n_in; (void)out_size; (void)ws_size;
    // inputs (setup_inputs order)
    const float* node_feat = (const float*)d_in[1];
    const int*   edge_idx  = (const int*)d_in[2];
    const int*   batch_idx = (const int*)d_in[3];
    const float* W0        = (const float*)d_in[4];
    const float* W_rest    = (const float*)d_in[5];
    const float* att_src   = (const float*)d_in[6];
    const float* att_dst   = (const float*)d_in[7];
    const float* gat_bias  = (const float*)d_in[8];
    const float* ln_gamma  = (const float*)d_in[9];
    const float* ln_beta   = (const float*)d_in[10];
    const float* proj_W1   = (const float*)d_in[11];
    const float* proj_b1   = (const float*)d_in[12];
    const float* proj_W2   = (const float*)d_in[13];
    const float* proj_b2   = (const float*)d_in[14];
    const float* out_gamma = (const float*)d_in[15];
    const float* out_beta  = (const float*)d_in[16];
    float* out = (float*)d_out;

    // workspace layout (floats)
    float* ws = (float*)d_ws;
    float*    h       = ws;                       // N*64
    float*    xh      = h      + (size_t)N_NODES * HID;      // N*64
    float*    acc     = xh     + (size_t)N_NODES * HID;      // N*64
    float*    alpha_s = acc    + (size_t)N_NODES * HID;      // N*4
    float*    alpha_d = alpha_s + (size_t)N_NODES * HEADS;   // N*4
    unsigned* emaxU   = (unsigned*)(alpha_d + (size_t)N_NODES * HEADS); // N*4
    float*    denom   = (float*)(emaxU + (size_t)N_NODES * HEADS);      // N*4
    float*    ezbuf   = denom  + (size_t)N_NODES * HEADS;    // E2*4
    float*    hsum    = ezbuf  + (size_t)E2 * HEADS;         // M*64
    float*    cnt     = hsum   + (size_t)N_MOLS * HID;       // M
    float*    hp      = cnt    + (size_t)N_MOLS;             // M*128
    float*    z1      = hp     + (size_t)N_MOLS * 2 * HID;   // M*64
    float*    z2      = z1     + (size_t)N_MOLS * HID;       // M*256

    const int T = 256;
    const int fillBlocks = 2048;

    for (int layer = 0; layer < 3; ++layer) {
        const int K = (layer == 0) ? 32 : HID;
        const float* A = (layer == 0) ? node_feat : h;
        const float* W = (layer == 0) ? W0 : (W_rest + (size_t)(layer - 1) * HID * HID);

        // xh = A @ W  (WMMA)
        {
            const int tiles = (N_NODES / 16) * (HID / 16);
            wmma_gemm_kernel<<<cdiv(tiles, 8), T, 0, stream>>>(
                A, W, nullptr, xh, N_NODES, K, HID, 0);
        }
        // attention logits
        att_alpha_kernel<<<cdiv((long)N_NODES * HEADS, T), T, 0, stream>>>(
            xh, att_src + (size_t)layer * HEADS * FH, att_dst + (size_t)layer * HEADS * FH,
            alpha_s, alpha_d);
        // init accumulators
        fill_u32_kernel<<<fillBlocks, T, 0, stream>>>(emaxU, ENC_NEG_INF, (long)N_NODES * HEADS);
        fill_f32_kernel<<<fillBlocks, T, 0, stream>>>(denom, 0.f, (long)N_NODES * HEADS);
        fill_f32_kernel<<<fillBlocks, T, 0, stream>>>(acc, 0.f, (long)N_NODES * HID);
        // segment softmax + message aggregation
        edge_max_kernel<<<cdiv((long)E2 * HEADS, T), T, 0, stream>>>(edge_idx, alpha_s, alpha_d, emaxU);
        edge_exp_kernel<<<cdiv((long)E2 * HEADS, T), T, 0, stream>>>(edge_idx, alpha_s, alpha_d, emaxU, ezbuf, denom);
        edge_msg_kernel<<<cdiv((long)E2 * HID, T), T, 0, stream>>>(edge_idx, xh, ezbuf, denom, acc);
        // LayerNorm(acc + bias) (+ relu for layers 0,1) -> h
        ln_node_kernel<<<cdiv(N_NODES, 8), T, 0, stream>>>(
            acc, gat_bias + (size_t)layer * HID, ln_gamma + (size_t)layer * HID,
            ln_beta + (size_t)layer * HID, h, layer < 2 ? 1 : 0);
    }

    // molecule pooling
    fill_f32_kernel<<<fillBlocks, T, 0, stream>>>(hsum, 0.f, (long)N_MOLS * HID);
    fill_f32_kernel<<<cdiv(N_MOLS, T), T, 0, stream>>>(cnt, 0.f, (long)N_MOLS);
    pool_kernel<<<cdiv((long)N_NODES * HID, T), T, 0, stream>>>(h, batch_idx, hsum, cnt);
    hp_kernel<<<cdiv((long)N_MOLS * 2 * HID, T), T, 0, stream>>>(hsum, cnt, hp);

    // MLP head (WMMA)
    {
        const int tiles1 = (N_MOLS / 16) * (HID / 16);
        wmma_gemm_kernel<<<cdiv(tiles1, 8), T, 0, stream>>>(
            hp, proj_W1, proj_b1, z1, N_MOLS, 2 * HID, HID, 1);
        const int tiles2 = (N_MOLS / 16) * (DOUT / 16);
        wmma_gemm_kernel<<<cdiv(tiles2, 8), T, 0, stream>>>(
            z1, proj_W2, proj_b2, z2, N_MOLS, HID, DOUT, 0);
    }

    // final LayerNorm -> output
    ln_out_kernel<<<cdiv(N_MOLS, 8), T, 0, stream>>>(z2, out_gamma, out_beta, out);
}